// BoWExtractor_46772193853838
// MI455X (gfx1250) — compile-verified
//
#include <hip/hip_runtime.h>
#include <hip/hip_bf16.h>

typedef float v2f __attribute__((ext_vector_type(2)));
typedef float v8f __attribute__((ext_vector_type(8)));

#define NB 32          // batch
#define NC 256         // channels (GEMM K)
#define NK 8192        // codebook entries (GEMM N)
#define NP 196         // 14*14 interior pixels per batch (GEMM M)
#define MT 16          // M tile (pixels per workgroup)
#define NWAVES 8
#define NBLOCK 128     // codes per block iteration (8 waves * 16)
#define NITER (NK / NBLOCK)
#define SCALE 30.0f    // INV_DELTA / MIN_DIST_MEAN
#define NEGBIG (-3.0e38f)

// One 16x16 (MxN) f32 tile accumulated over K=256 with V_WMMA_F32_16X16X4_F32.
// Erow = E + n*256 + ca   (ca = 0 for lanes 0-15, 2 for lanes 16-31)
// Acol = &A_lds[ca*16 + (lane&15)]   (A_lds is [c][m], c-major stride 16)
__device__ __forceinline__ v8f tile_mma(const float* __restrict__ Erow,
                                        const float* __restrict__ Acol) {
  v8f acc = {};
#pragma unroll 4
  for (int c = 0; c < NC; c += 4) {
    v2f a, bb;
    a.x  = Acol[(c + 0) * MT];   // A: K = c+ca+0
    a.y  = Acol[(c + 1) * MT];   // A: K = c+ca+1
    bb.x = Erow[c + 0];          // B: K = c+ca+0 (contiguous -> b64 load)
    bb.y = Erow[c + 1];
    acc = __builtin_amdgcn_wmma_f32_16x16x4_f32(
        /*neg_a=*/false, a, /*neg_b=*/false, bb,
        /*c_mod=*/(short)0, acc, /*reuse_a=*/false, /*reuse_b=*/false);
  }
  return acc;
}

__global__ void __launch_bounds__(256) esq_kernel(const float* __restrict__ E,
                                                  float* __restrict__ esq) {
  int k = blockIdx.x * 256 + threadIdx.x;
  if (k >= NK) return;
  const float4* row = (const float4*)(E + (size_t)k * NC);
  float s = 0.f;
#pragma unroll 4
  for (int c = 0; c < NC / 4; ++c) {
    float4 v = row[c];
    s += v.x * v.x + v.y * v.y + v.z * v.z + v.w * v.w;
  }
  esq[k] = s;
}

__global__ void __launch_bounds__(256) init_bow(unsigned* __restrict__ bow) {
  int i = blockIdx.x * 256 + threadIdx.x;
  if (i < NB * NK) bow[i] = 0u;
}

__global__ void __launch_bounds__(256) bow_main(
    const float* __restrict__ F, const float* __restrict__ E,
    const float* __restrict__ esq, float* __restrict__ codes,
    unsigned* __restrict__ bow) {
  __shared__ float A_lds[NC * MT];                 // 16 KB, [c][m]
  __shared__ float fsq[MT];
  __shared__ float Mfin[MT];
  __shared__ float Sinv[MT];
  __shared__ float red[NWAVES * MT * 16 * 2];      // 16 KB, per-(wave,m,nlane) {max,sum}
  __shared__ float tbuf[NWAVES][16 * 17];          // ~8.5 KB transpose staging

  const int b    = blockIdx.y;
  const int m0   = blockIdx.x * MT;
  const int tid  = threadIdx.x;
  const int wave = tid >> 5;
  const int lane = tid & 31;
  const int half = lane >> 4;     // 0 | 1
  const int nl   = lane & 15;
  const int ca   = half * 2;      // K sub-offset for this lane half
  const int moff = half * 8;      // M offset of this lane half in C layout

  // ---- Stage A tile (features) into LDS in WMMA A-friendly [c][m] layout ----
#pragma unroll
  for (int it = 0; it < 16; ++it) {
    int c = it * 16 + (tid >> 4);
    int m = tid & 15;
    int p = m0 + m; if (p > NP - 1) p = NP - 1;     // clamp pad pixels
    int h = p / 14 + 1, w = p % 14 + 1;             // interior crop
    A_lds[c * MT + m] = F[(((size_t)b * NC + c) * 16 + h) * 16 + w];
  }
  __syncthreads();
  if (tid < MT) {
    float s = 0.f;
    for (int c = 0; c < NC; ++c) { float v = A_lds[c * MT + tid]; s += v * v; }
    fsq[tid] = s;
  }
  __syncthreads();

  float fs[8];
#pragma unroll
  for (int v = 0; v < 8; ++v) fs[v] = fsq[v + moff];

  const float* Acol = A_lds + ca * MT + nl;

  // ---- Pass 1: online softmax statistics over all 8192 codes ----
  float mx[8], sm[8];
#pragma unroll
  for (int v = 0; v < 8; ++v) { mx[v] = NEGBIG; sm[v] = 0.f; }

  for (int it = 0; it < NITER; ++it) {
    int n = it * NBLOCK + wave * 16 + nl;
    const float* Erow = E + (size_t)n * NC + ca;
    if (it + 1 < NITER)
      __builtin_prefetch(Erow + (size_t)NBLOCK * NC, 0, 0);  // global_prefetch_b8
    v8f acc = tile_mma(Erow, Acol);
    float eq = esq[n];
#pragma unroll
    for (int v = 0; v < 8; ++v) {
      float d = fs[v] - 2.f * acc[v] + eq;
      float s = -SCALE * d;
      float nm = fmaxf(mx[v], s);
      sm[v] = sm[v] * __expf(mx[v] - nm) + __expf(s - nm);
      mx[v] = nm;
    }
  }
#pragma unroll
  for (int v = 0; v < 8; ++v) {
    int m = v + moff;
    int idx = ((wave * MT + m) * 16 + nl) * 2;
    red[idx] = mx[v]; red[idx + 1] = sm[v];
  }
  __syncthreads();
  if (tid < MT) {                                  // merge 8 waves x 16 lanes
    float M = NEGBIG, S = 0.f;
    for (int w8 = 0; w8 < NWAVES; ++w8)
      for (int j = 0; j < 16; ++j) {
        int idx = ((w8 * MT + tid) * 16 + j) * 2;
        float m2 = red[idx], s2 = red[idx + 1];
        float nm = fmaxf(M, m2);
        S = S * __expf(M - nm) + s2 * __expf(m2 - nm);
        M = nm;
      }
    Mfin[tid] = M;
    Sinv[tid] = 1.f / S;
  }
  __syncthreads();

  float Mv[8], Si[8];
#pragma unroll
  for (int v = 0; v < 8; ++v) { Mv[v] = Mfin[v + moff]; Si[v] = Sinv[v + moff]; }

  // ---- Pass 2: recompute, write codes (coalesced via LDS transpose) + bow ----
  for (int it = 0; it < NITER; ++it) {
    int n0 = it * NBLOCK + wave * 16;
    int n  = n0 + nl;
    const float* Erow = E + (size_t)n * NC + ca;
    if (it + 1 < NITER)
      __builtin_prefetch(Erow + (size_t)NBLOCK * NC, 0, 0);
    v8f acc = tile_mma(Erow, Acol);
    float eq = esq[n];
#pragma unroll
    for (int v = 0; v < 8; ++v) {
      float d = fs[v] - 2.f * acc[v] + eq;
      float s = -SCALE * d;
      float code = __expf(s - Mv[v]) * Si[v];
      int m = v + moff;
      tbuf[wave][nl * 17 + m] = (m0 + m < NP) ? code : 0.f;  // pad -> 0 (safe for max)
    }
    __syncthreads();
    {
      // stores: lane -> pixel m = nl (contiguous 64B runs per n)
      int m = nl;
      int p = m0 + m;
#pragma unroll
      for (int j = 0; j < 8; ++j) {
        int nr = j * 2 + half;
        if (p < NP)
          codes[((size_t)b * NK + (size_t)(n0 + nr)) * NP + p] = tbuf[wave][nr * 17 + m];
      }
      // bow candidate: max over this tile's 16 pixels, monotone uint atomic
      if (half == 0) {
        float mxv = 0.f;
#pragma unroll
        for (int mm = 0; mm < 16; ++mm) mxv = fmaxf(mxv, tbuf[wave][nl * 17 + mm]);
        atomicMax(&bow[(size_t)b * NK + n0 + nl], __float_as_uint(mxv));
      }
    }
    __syncthreads();
  }
}

__global__ void __launch_bounds__(256) bow_norm(const unsigned* __restrict__ bow_ws,
                                                float* __restrict__ out_bow) {
  __shared__ float part[256];
  __shared__ float invTot;
  int b = blockIdx.x;
  float s = 0.f;
  for (int k = threadIdx.x; k < NK; k += 256)
    s += __uint_as_float(bow_ws[(size_t)b * NK + k]);
  part[threadIdx.x] = s;
  __syncthreads();
  for (int st = 128; st > 0; st >>= 1) {
    if (threadIdx.x < st) part[threadIdx.x] += part[threadIdx.x + st];
    __syncthreads();
  }
  if (threadIdx.x == 0) invTot = 1.f / fmaxf(part[0], 1e-12f);
  __syncthreads();
  float inv = invTot;
  for (int k = threadIdx.x; k < NK; k += 256)
    out_bow[(size_t)b * NK + k] = __uint_as_float(bow_ws[(size_t)b * NK + k]) * inv;
}

extern "C" void kernel_launch(void* const* d_in, const int* in_sizes, int n_in,
                              void* d_out, int out_size, void* d_ws, size_t ws_size,
                              hipStream_t stream) {
  const float* F = (const float*)d_in[0];   // (32,256,16,16)
  const float* E = (const float*)d_in[1];   // (8192,256), already ReLU'd by setup
  float* out       = (float*)d_out;
  float* out_bow   = out;                   // 32*8192
  float* out_codes = out + (size_t)NB * NK; // 32*8192*196

  float*    esq    = (float*)d_ws;                                    // 8192 f32
  unsigned* bow_ws = (unsigned*)((char*)d_ws + NK * sizeof(float));   // 32*8192 u32

  init_bow<<<(NB * NK + 255) / 256, 256, 0, stream>>>(bow_ws);
  esq_kernel<<<NK / 256, 256, 0, stream>>>(E, esq);
  bow_main<<<dim3((NP + MT - 1) / MT, NB), 256, 0, stream>>>(F, E, esq, out_codes, bow_ws);
  bow_norm<<<NB, 256, 0, stream>>>(bow_ws, out_bow);
}